// IPM_13400297963831
// MI455X (gfx1250) — compile-verified
//
#include <hip/hip_runtime.h>
#include <hip/hip_bf16.h>
#include <math.h>

typedef float v2f __attribute__((ext_vector_type(2)));
typedef float v8f __attribute__((ext_vector_type(8)));

#define B_   2
#define N_   6
#define C_   32
#define H_   128
#define W_   352
#define TH_  400
#define TW_  400
#define HW_  (H_ * W_)
#define EPS_ 1e-7f

// Build 4x4 rotation (row-major) from Euler angles in degrees, exactly the
// reference formula (si=roll, sj=pitch, sk=yaw).
__device__ inline void euler_rot(float roll, float pitch, float yaw, float R[16]) {
    const float d2r = 0.017453292519943295f;
    float si = sinf(roll * d2r),  sj = sinf(pitch * d2r), sk = sinf(yaw * d2r);
    float ci = cosf(roll * d2r),  cj = cosf(pitch * d2r), ck = cosf(yaw * d2r);
    float cc = ci * ck, cs = ci * sk, sc = si * ck, ss = si * sk;
    R[0]  = cj * ck; R[1]  = sj * sc - cs; R[2]  = sj * cc + ss; R[3]  = 0.f;
    R[4]  = cj * sk; R[5]  = sj * ss + cc; R[6]  = sj * cs - sc; R[7]  = 0.f;
    R[8]  = -sj;     R[9]  = cj * si;      R[10] = cj * ci;      R[11] = 0.f;
    R[12] = 0.f;     R[13] = 0.f;          R[14] = 0.f;          R[15] = 1.f;
}

// One wave32. Computes M[idx] = (Ks[idx] @ RTs[idx]) @ R_euler[idx % B] for
// idx = b*N+n (the reference's jnp.tile makes camera (b,n) use plane batch
// (b*N+n)%B), then folds z into the translation column. 4 independent 4x4
// products are packed block-diagonally into each V_WMMA_F32_16X16X4_F32:
//   A rows 4g+i = K_g row i  (A 16x4 layout: lane=row, lane-half = K pair)
//   B cols 4g+j = RT_g col j (B 4x16 layout: lane-half = upper K half)
// Diagonal 4x4 blocks of D are the products; staged through LDS between the
// two WMMA rounds. Emits 6 v_wmma_f32_16x16x4_f32.
__global__ __launch_bounds__(32)
void ipm_setup(const float* __restrict__ Ks,   const float* __restrict__ RTs,
               const float* __restrict__ zs,   const float* __restrict__ yaws,
               const float* __restrict__ rolls,const float* __restrict__ pitchs,
               float* __restrict__ geo) {
    __shared__ float Pbuf[B_ * N_ * 16];
    __shared__ float Mbuf[B_ * N_ * 16];

    const int l    = threadIdx.x;   // 0..31
    const int half = l >> 4;        // which K pair this lane supplies
    const int q    = l & 15;        // A: packed row;  B: packed column
    const int grp  = q >> 2;        // which 4x4 block (matrix) in the pack
    const int sub  = q & 3;         // row-in-block (A) / col-in-block (B)
    const int kb   = half * 2;      // first K index held by this lane

    // ---- Stage 1: P = K @ RT (3 WMMAs cover 12 matrices) ----
    for (int base = 0; base < B_ * N_; base += 4) {
        int idx = base + grp;
        v2f a, bm;
        a.x  = Ks[idx * 16 + sub * 4 + kb];        // K_idx[sub][kb]
        a.y  = Ks[idx * 16 + sub * 4 + kb + 1];    // K_idx[sub][kb+1]
        bm.x = RTs[idx * 16 + kb * 4 + sub];       // RT_idx[kb][sub]
        bm.y = RTs[idx * 16 + (kb + 1) * 4 + sub]; // RT_idx[kb+1][sub]
        v8f c = {};
        v8f d = __builtin_amdgcn_wmma_f32_16x16x4_f32(
            false, a, false, bm, (short)0, c, false, false);
        // D layout: lane<16 holds rows 0-7 of column q; lane>=16 rows 8-15.
        // Keep only diagonal 4x4 blocks.
        #pragma unroll
        for (int v = 0; v < 8; ++v) {
            int r = v + half * 8;
            if ((r >> 2) == (q >> 2)) {
                Pbuf[(base + (r >> 2)) * 16 + (r & 3) * 4 + sub] = d[v];
            }
        }
    }
    __syncthreads();

    // ---- Stage 2: M = P @ R_euler[idx % B] ----
    for (int base = 0; base < B_ * N_; base += 4) {
        int idx = base + grp;
        int pb  = idx % B_;                // plane-batch quirk from jnp.tile
        float R[16];
        euler_rot(rolls[pb], pitchs[pb], yaws[pb], R);
        v2f a, bm;
        a.x  = Pbuf[idx * 16 + sub * 4 + kb];
        a.y  = Pbuf[idx * 16 + sub * 4 + kb + 1];
        bm.x = R[kb * 4 + sub];
        bm.y = R[(kb + 1) * 4 + sub];
        v8f c = {};
        v8f d = __builtin_amdgcn_wmma_f32_16x16x4_f32(
            false, a, false, bm, (short)0, c, false, false);
        #pragma unroll
        for (int v = 0; v < 8; ++v) {
            int r = v + half * 8;
            if ((r >> 2) == (q >> 2)) {
                Mbuf[(base + (r >> 2)) * 16 + (r & 3) * 4 + sub] = d[v];
            }
        }
    }
    __syncthreads();

    // ---- Fold z: pix_row_r = M[r][0]*yy + M[r][1]*xx + (M[r][2]*z + M[r][3])
    if (l < B_ * N_) {
        int   idx = l;
        int   pb  = idx % B_;
        float z   = zs[pb];
        #pragma unroll
        for (int r = 0; r < 3; ++r) {
            geo[idx * 9 + r * 3 + 0] = Mbuf[idx * 16 + r * 4 + 0];
            geo[idx * 9 + r * 3 + 1] = Mbuf[idx * 16 + r * 4 + 1];
            geo[idx * 9 + r * 3 + 2] = Mbuf[idx * 16 + r * 4 + 2] * z
                                     + Mbuf[idx * 16 + r * 4 + 3];
        }
    }
}

// One thread per output point (b, ti, tj). Lanes are tj-contiguous so corner
// gathers and per-channel stores are coalesced. 32 channel max-accumulators
// in registers; images are L2-resident (69 MB < 192 MB L2).
__global__ __launch_bounds__(256)
void ipm_main(const float* __restrict__ images,
              const float* __restrict__ geo,
              float* __restrict__ out) {
    const int total = B_ * TH_ * TW_;
    int tid = blockIdx.x * blockDim.x + threadIdx.x;
    if (tid >= total) return;

    int b   = tid / (TH_ * TW_);
    int rem = tid - b * (TH_ * TW_);
    int ti  = rem / TW_;
    int tj  = rem - ti * TW_;

    const float step = 100.0f / 399.0f;         // linspace(-50,50,400)
    float yy = -50.0f + ti * step;              // varies with row (ti)
    float xx = -50.0f + tj * step;              // varies with col (tj)

    float acc[C_];
    #pragma unroll
    for (int c = 0; c < C_; ++c) acc[c] = -INFINITY;

    #pragma unroll 1
    for (int n = 0; n < N_; ++n) {
        const float* g = geo + (b * N_ + n) * 9;
        float hx = g[0] * yy + g[1] * xx + g[2];
        float hy = g[3] * yy + g[4] * xx + g[5];
        float hw = g[6] * yy + g[7] * xx + g[8];
        float denom = hw + EPS_;
        float px = hx / denom;                  // reference: pix / (w + eps)
        float py = hy / denom;

        float fx = floorf(px), fy = floorf(py);
        float x0 = fminf(fmaxf(fx,        0.0f), (float)(W_ - 1));
        float x1 = fminf(fmaxf(fx + 1.0f, 0.0f), (float)(W_ - 1));
        float y0 = fminf(fmaxf(fy,        0.0f), (float)(H_ - 1));
        float y1 = fminf(fmaxf(fy + 1.0f, 0.0f), (float)(H_ - 1));
        // weights from CLIPPED corners, exactly as the reference
        float wx0 = x1 - px, wx1 = px - x0;
        float wy0 = y1 - py, wy1 = py - y0;
        float w00 = wx0 * wy0, w01 = wx0 * wy1;
        float w10 = wx1 * wy0, w11 = wx1 * wy1;

        int ix0 = (int)x0, ix1 = (int)x1, iy0 = (int)y0, iy1 = (int)y1;
        int i00 = iy0 * W_ + ix0;               // (x0,y0) -> w00
        int i01 = iy1 * W_ + ix0;               // (x0,y1) -> w01
        int i10 = iy0 * W_ + ix1;               // (x1,y0) -> w10
        int i11 = iy1 * W_ + ix1;               // (x1,y1) -> w11

        const float* base = images + ((size_t)(b * N_ + n)) * C_ * HW_;
        #pragma unroll
        for (int c = 0; c < C_; ++c) {
            const float* pc = base + (size_t)c * HW_;
            float v = w00 * pc[i00] + w01 * pc[i01]
                    + w10 * pc[i10] + w11 * pc[i11];
            acc[c] = fmaxf(acc[c], v);
        }
    }

    float* ob = out + ((size_t)b * C_) * (TH_ * TW_) + ti * TW_ + tj;
    #pragma unroll
    for (int c = 0; c < C_; ++c)
        ob[(size_t)c * (TH_ * TW_)] = acc[c];
}

extern "C" void kernel_launch(void* const* d_in, const int* in_sizes, int n_in,
                              void* d_out, int out_size, void* d_ws, size_t ws_size,
                              hipStream_t stream) {
    const float* images = (const float*)d_in[0];
    const float* Ks     = (const float*)d_in[1];
    const float* RTs    = (const float*)d_in[2];
    const float* zs     = (const float*)d_in[3];
    const float* yaws   = (const float*)d_in[4];
    const float* rolls  = (const float*)d_in[5];
    const float* pitchs = (const float*)d_in[6];
    float* out = (float*)d_out;
    float* geo = (float*)d_ws;   // 12 cameras x 9 coefficients = 432 B

    hipLaunchKernelGGL(ipm_setup, dim3(1), dim3(32), 0, stream,
                       Ks, RTs, zs, yaws, rolls, pitchs, geo);

    const int total   = B_ * TH_ * TW_;
    const int threads = 256;
    hipLaunchKernelGGL(ipm_main, dim3((total + threads - 1) / threads),
                       dim3(threads), 0, stream, images, geo, out);
}